// Decoder_BCD_5823975654050
// MI455X (gfx1250) — compile-verified
//
#include <hip/hip_runtime.h>
#include <hip/hip_bf16.h>
#include <math.h>

// ---------------------------------------------------------------------------
// Problem constants (from reference)
// ---------------------------------------------------------------------------
#define DIMX 128
#define LDIM 8128           // 128*127/2
#define NB   32             // batch B
#define NS   2048           // samples S
#define HID  512
#define NN3  16384          // DIM*DIM
#define PROJ 256

#define LOG_SIG   (-2.3025850929940455f)   // log(0.1)
#define HALF_L2PI (0.9189385332046727f)    // 0.5*log(2*pi)

typedef __attribute__((ext_vector_type(2))) float v2f;
typedef __attribute__((ext_vector_type(8))) float v8f;

// ---------------------------------------------------------------------------
// CDNA5 WMMA: D(16x16,f32) = A(16x4,f32) x B(4x16,f32) + C   (exact fp32)
// A frag (ISA 7.12.2): lane = 16*h + m ; a[r] = A[m, k0 + r + 2h]
// B frag (mirror)    : lane = 16*h + n ; b[r] = B[k0 + r + 2h, n]
// C/D                : lane = 16*h + n ; c[r] = C[r + 8h, n]
// ---------------------------------------------------------------------------
__device__ __forceinline__ v8f wmma4(v2f a, v2f b, v8f c) {
#if __has_builtin(__builtin_amdgcn_wmma_f32_16x16x4_f32)
  return __builtin_amdgcn_wmma_f32_16x16x4_f32(false, a, false, b,
                                               (short)0, c, false, false);
#else
  // lane-shuffle emulation (keeps TU compiling if builtin is absent)
  const int lane = threadIdx.x & 31;
  const int n = lane & 15;
#pragma unroll
  for (int r = 0; r < 8; ++r) {
    int m = r + 8 * (lane >> 4);
    float acc = c[r];
#pragma unroll
    for (int kk = 0; kk < 4; ++kk) {
      float av = __shfl((kk & 1) ? a.y : a.x, m + 16 * (kk >> 1), 32);
      float bv = __shfl((kk & 1) ? b.y : b.x, n + 16 * (kk >> 1), 32);
      acc = fmaf(av, bv, acc);
    }
    c[r] = acc;
  }
  return c;
#endif
}

__device__ __forceinline__ v8f zero8() {
  v8f c;
#pragma unroll
  for (int r = 0; r < 8; ++r) c[r] = 0.f;
  return c;
}

__device__ __forceinline__ v2f load_a(const float* A, int lda, int m0, int k, int lane) {
  const int h = lane >> 4, m = lane & 15;
  const float* p = A + (size_t)(m0 + m) * lda + k + 2 * h;
  v2f a; a.x = p[0]; a.y = p[1];
  return a;
}
__device__ __forceinline__ v2f load_b(const float* B, int ldb, int n0, int k, int lane) {
  const int h = lane >> 4, n = lane & 15;
  v2f b;
  b.x = B[(size_t)(k + 2 * h)     * ldb + n0 + n];
  b.y = B[(size_t)(k + 2 * h + 1) * ldb + n0 + n];
  return b;
}

__device__ __forceinline__ float gelu_tanh(float x) {
  float x3 = x * x * x;
  return 0.5f * x * (1.f + tanhf(0.7978845608028654f * (x + 0.044715f * x3)));
}

// ---------------------------------------------------------------------------
// K0: init log-noises, last column of full_l_batch, zero log-prob
// ---------------------------------------------------------------------------
__global__ void k_init(float* logn, float* flp, float* flb) {
  int t = blockIdx.x * 256 + threadIdx.x;
  if (t < NB * DIMX) logn[t] = LOG_SIG;
  if (t < NB) { flp[t] = 0.f; flb[(size_t)t * (LDIM + 1) + LDIM] = LOG_SIG; }
}

// ---------------------------------------------------------------------------
// K1: l_batch = mean + exp(tanh-clamped log_std)*z ; partial log-prob sums
// ---------------------------------------------------------------------------
__global__ void k_lbatch(const float* Lp, const float* z_l, float* flb, float* part) {
  __shared__ float red[256];
  const int b = blockIdx.y;
  const int k = blockIdx.x * 256 + threadIdx.x;
  float contrib = 0.f;
  if (k < LDIM) {
    float mean = Lp[k];
    float ls = tanhf(Lp[LDIM + k] * 0.1f) * 10.f;
    float z = z_l[(size_t)b * LDIM + k];
    flb[(size_t)b * (LDIM + 1) + k] = fmaf(expf(ls), z, mean);
    contrib = -0.5f * z * z - ls - HALF_L2PI;
  }
  red[threadIdx.x] = contrib;
  __syncthreads();
  for (int off = 128; off > 0; off >>= 1) {
    if (threadIdx.x < off) red[threadIdx.x] += red[threadIdx.x + off];
    __syncthreads();
  }
  if (threadIdx.x == 0) part[b * 32 + blockIdx.x] = red[0];
}

__global__ void k_lp_reduce(const float* part, float* flp) {
  int b = threadIdx.x;  // 32 threads
  float s = 0.f;
  for (int i = 0; i < 32; ++i) s += part[b * 32 + i];  // fixed order: deterministic
  flp[b] = s;
}

// ---------------------------------------------------------------------------
// K2: batched_L[b,i,j] = (i>j) ? l_batch[b, triu_k(j,i)] : 0   (strict lower)
// ---------------------------------------------------------------------------
__global__ void k_buildL(const float* flb, float* Lout) {
  const int b = blockIdx.y;
  const int idx = blockIdx.x * 256 + threadIdx.x;  // 0..16383
  const int i = idx >> 7, j = idx & 127;
  float v = 0.f;
  if (i > j) {
    int kk = j * 127 - (j * (j - 1)) / 2 + (i - j - 1);
    v = flb[(size_t)b * (LDIM + 1) + kk];
  }
  Lout[(size_t)b * NN3 + idx] = v;
}

// ---------------------------------------------------------------------------
// K3: h1 = gelu(full_l_batch @ W1 + b1), last-K (log_sig col) folded into bias
//     grid (32 Ntiles, 2 Mtiles), 1 wave / block, WMMA fp32 16x16x4
// ---------------------------------------------------------------------------
__global__ void k_gemm1(const float* flb, const float* W1, const float* b1, float* h1) {
  const int lane = threadIdx.x;
  const int n0 = blockIdx.x * 16, m0 = blockIdx.y * 16;
  v8f c = zero8();
  for (int k = 0; k < LDIM; k += 4) {
    v2f a = load_a(flb, LDIM + 1, m0, k, lane);
    v2f b = load_b(W1, HID, n0, k, lane);
    c = wmma4(a, b, c);
  }
  const int h = lane >> 4, nl = lane & 15, col = n0 + nl;
  const float bias = b1[col] + LOG_SIG * W1[(size_t)LDIM * HID + col];
#pragma unroll
  for (int r = 0; r < 8; ++r)
    h1[(size_t)(m0 + r + 8 * h) * HID + col] = gelu_tanh(c[r] + bias);
}

// K4: h2 = gelu(h1 @ W2 + b2)
__global__ void k_gemm2(const float* h1, const float* W2, const float* b2, float* h2) {
  const int lane = threadIdx.x;
  const int n0 = blockIdx.x * 16, m0 = blockIdx.y * 16;
  v8f c = zero8();
  for (int k = 0; k < HID; k += 4) {
    v2f a = load_a(h1, HID, m0, k, lane);
    v2f b = load_b(W2, HID, n0, k, lane);
    c = wmma4(a, b, c);
  }
  const int h = lane >> 4, nl = lane & 15, col = n0 + nl;
  const float bias = b2[col];
#pragma unroll
  for (int r = 0; r < 8; ++r)
    h2[(size_t)(m0 + r + 8 * h) * HID + col] = gelu_tanh(c[r] + bias);
}

// ---------------------------------------------------------------------------
// K5: p_logits = tanh((h2@W3+b3)/10)*10 -> out ; la = p_logits + gumbel (TAU=1)
//     one wave per N-tile, both M-tiles share the B loads
// ---------------------------------------------------------------------------
__global__ void k_gemm3(const float* h2, const float* W3, const float* b3,
                        const float* gum, float* out_logits, float* la) {
  const int lane = threadIdx.x;
  const int n0 = blockIdx.x * 16;
  v8f c0 = zero8(), c1 = zero8();
  for (int k = 0; k < HID; k += 4) {
    if (k + 4 < HID) __builtin_prefetch(&W3[(size_t)(k + 4) * NN3 + n0], 0, 1);
    v2f a0 = load_a(h2, HID, 0, k, lane);
    v2f a1 = load_a(h2, HID, 16, k, lane);
    v2f b = load_b(W3, NN3, n0, k, lane);
    c0 = wmma4(a0, b, c0);
    c1 = wmma4(a1, b, c1);
  }
  const int h = lane >> 4, nl = lane & 15, col = n0 + nl;
  const float bias = b3[col];
#pragma unroll
  for (int r = 0; r < 8; ++r) {
#pragma unroll
    for (int t = 0; t < 2; ++t) {
      int row = 16 * t + r + 8 * h;  // row == batch index
      float x = tanhf((((t == 0) ? c0[r] : c1[r]) + bias) * 0.1f) * 10.f;
      size_t o = (size_t)row * NN3 + col;
      out_logits[o] = x;
      float u = gum[o];
      u = fminf(fmaxf(u, 1e-6f), 1.f - 1e-6f);
      la[o] = x - logf(-logf(u));  // + gumbel, TAU = 1
    }
  }
}

// ---------------------------------------------------------------------------
// K6: 20 Sinkhorn iterations (row lse then col lse) + greedy hard columns.
//     One block per batch, 128 threads, LDS matrix padded to stride 129.
// ---------------------------------------------------------------------------
__global__ void k_sinkhorn(const float* la_in, int* pc, int* order) {
  extern __shared__ float smem[];
  float* la = smem;                         // 128*129
  float* redv = la + 128 * 129;             // 128
  int* redi = (int*)(redv + 128);           // 128
  int* used = redi + 128;                   // 128
  int* pcl = used + 128;                    // 128
  const int b = blockIdx.x, tid = threadIdx.x;

  for (int i = tid; i < NN3; i += 128)
    la[(i >> 7) * 129 + (i & 127)] = la_in[(size_t)b * NN3 + i];
  used[tid] = 0;
  __syncthreads();

  for (int it = 0; it < 20; ++it) {
    {  // row pass: tid = row
      float m = -INFINITY;
      for (int cc = 0; cc < 128; ++cc) m = fmaxf(m, la[tid * 129 + cc]);
      float s = 0.f;
      for (int cc = 0; cc < 128; ++cc) s += expf(la[tid * 129 + cc] - m);
      float lse = m + logf(s);
      for (int cc = 0; cc < 128; ++cc) la[tid * 129 + cc] -= lse;
    }
    __syncthreads();
    {  // col pass: tid = col
      float m = -INFINITY;
      for (int rr = 0; rr < 128; ++rr) m = fmaxf(m, la[rr * 129 + tid]);
      float s = 0.f;
      for (int rr = 0; rr < 128; ++rr) s += expf(la[rr * 129 + tid] - m);
      float lse = m + logf(s);
      for (int rr = 0; rr < 128; ++rr) la[rr * 129 + tid] -= lse;
    }
    __syncthreads();
  }

  // greedy: per row, argmax over unused columns (first-max tie-break)
  for (int row = 0; row < 128; ++row) {
    redv[tid] = used[tid] ? -INFINITY : la[row * 129 + tid];
    redi[tid] = tid;
    __syncthreads();
    for (int off = 64; off > 0; off >>= 1) {
      if (tid < off) {
        float v2 = redv[tid + off], v1 = redv[tid];
        int i2 = redi[tid + off], i1 = redi[tid];
        if (v2 > v1 || (v2 == v1 && i2 < i1)) { redv[tid] = v2; redi[tid] = i2; }
      }
      __syncthreads();
    }
    if (tid == 0) { pcl[row] = redi[0]; used[redi[0]] = 1; }
    __syncthreads();
  }
  pc[b * 128 + tid] = pcl[tid];
  order[b * 128 + pcl[tid]] = tid;  // inverse permutation
}

// ---------------------------------------------------------------------------
// K7: batched_P = one_hot(pc) ; batched_W[b,i,j] = L[b, pc[j], pc[i]]
// ---------------------------------------------------------------------------
__global__ void k_permW(const int* pc, const float* Lm, float* P, float* W) {
  const int b = blockIdx.y, i = blockIdx.x, j = threadIdx.x;
  const int pci = pc[b * 128 + i], pcj = pc[b * 128 + j];
  P[(size_t)b * NN3 + i * 128 + j] = (j == pci) ? 1.f : 0.f;
  W[(size_t)b * NN3 + i * 128 + j] = Lm[(size_t)b * NN3 + pcj * 128 + pci];
}

// ---------------------------------------------------------------------------
// K8: ancestral sampling == forward substitution y = L_b y + 0.1*z_anc[order]
//     in topological coordinates; intervention forces y[pc[it]] = 0.
//     One block = 128 samples of one batch; L_b broadcast from LDS.
// ---------------------------------------------------------------------------
__global__ void k_ancestral(const float* Lm, const int* ord, const int* pc,
                            const int* interv, const float* z_anc, float* qz) {
  extern __shared__ float smem[];
  float* Ll = smem;                      // 128*128
  float* Y = Ll + NN3;                   // 128 threads * stride 129 (no conflicts)
  int* ol = (int*)(Y + 128 * 129);       // 128
  const int b = blockIdx.y, tid = threadIdx.x;
  const int s = blockIdx.x * 128 + tid;

  for (int i = tid; i < NN3; i += 128) Ll[i] = Lm[(size_t)b * NN3 + i];
  ol[tid] = ord[b * 128 + tid];
  __syncthreads();

  const int it = interv[s];
  const int tz = (it < DIMX) ? pc[b * 128 + it] : -1;
  float* y = Y + tid * 129;
  const float* zs = z_anc + (size_t)s * DIMX;
  float* qrow = qz + ((size_t)b * NS + s) * DIMX;

  for (int t = 0; t < DIMX; ++t) {
    const float* Lrow = Ll + t * DIMX;   // same addr across lanes -> broadcast
    float acc = 0.f;
    for (int tp = 0; tp < t; ++tp) acc = fmaf(Lrow[tp], y[tp], acc);
    int od = ol[t];
    float val = (t == tz) ? 0.f : fmaf(0.1f, zs[od], acc);
    y[t] = val;
    qrow[od] = val;
  }
}

// ---------------------------------------------------------------------------
// K9: X_recons = qz(65536x128) @ W_dec(128x256), fp32 WMMA, 8 waves/block
// ---------------------------------------------------------------------------
__global__ void k_xrecons(const float* qz, const float* Wd, float* out) {
  const int lane = threadIdx.x;
  const int n0 = (blockIdx.x * 8 + threadIdx.y) * 16;  // 16 N-tiles
  const int m0 = blockIdx.y * 16;                      // 4096 M-tiles
  v8f c = zero8();
  const int h = lane >> 4, ml = lane & 15;
  const float* arow = qz + (size_t)(m0 + ml) * DIMX + 2 * h;
  for (int k = 0; k < DIMX; k += 4) {
    v2f a; a.x = arow[k]; a.y = arow[k + 1];           // 8B-aligned pair
    v2f b = load_b(Wd, PROJ, n0, k, lane);
    c = wmma4(a, b, c);
  }
  const int col = n0 + ml;
#pragma unroll
  for (int r = 0; r < 8; ++r)
    out[(size_t)(m0 + r + 8 * h) * PROJ + col] = c[r];
}

// ---------------------------------------------------------------------------
// Host launcher
// ---------------------------------------------------------------------------
extern "C" void kernel_launch(void* const* d_in, const int* in_sizes, int n_in,
                              void* d_out, int out_size, void* d_ws, size_t ws_size,
                              hipStream_t stream) {
  const int*   interv = (const int*)  d_in[0];
  const float* Lp     = (const float*)d_in[1];
  const float* z_l    = (const float*)d_in[2];
  const float* gum    = (const float*)d_in[3];
  const float* z_anc  = (const float*)d_in[4];
  const float* W1     = (const float*)d_in[5];
  const float* b1     = (const float*)d_in[6];
  const float* W2     = (const float*)d_in[7];
  const float* b2     = (const float*)d_in[8];
  const float* W3     = (const float*)d_in[9];
  const float* b3     = (const float*)d_in[10];
  const float* Wd     = (const float*)d_in[11];

  float* out = (float*)d_out;   // outputs concatenated flat, return order:
  float* o_P   = out;                     // (B,128,128)
  float* o_PL  = out + 524288;            // (B,128,128)
  float* o_L   = out + 1048576;           // (B,128,128)
  float* o_LN  = out + 1572864;           // (B,128)
  float* o_W   = out + 1576960;           // (B,128,128)
  float* o_QZ  = out + 2101248;           // (B,S,128)
  float* o_FLB = out + 10489856;          // (B,8129)
  float* o_FLP = out + 10749984;          // (B,)
  float* o_XR  = out + 10750016;          // (B,S,256)

  float* ws   = (float*)d_ws;
  float* w_h1 = ws;                        // 32*512
  float* w_h2 = ws + 16384;                // 32*512
  float* w_la = ws + 32768;                // 32*128*128
  int*   w_pc = (int*)(ws + 32768 + 524288);  // B*128
  int*   w_or = w_pc + 4096;                  // B*128
  float* w_pt = (float*)(w_or + 4096);        // 32*32 partials

  k_init     <<<16, 256, 0, stream>>>(o_LN, o_FLP, o_FLB);
  k_lbatch   <<<dim3(32, 32), 256, 0, stream>>>(Lp, z_l, o_FLB, w_pt);
  k_lp_reduce<<<1, 32, 0, stream>>>(w_pt, o_FLP);
  k_buildL   <<<dim3(64, 32), 256, 0, stream>>>(o_FLB, o_L);
  k_gemm1    <<<dim3(32, 2), 32, 0, stream>>>(o_FLB, W1, b1, w_h1);
  k_gemm2    <<<dim3(32, 2), 32, 0, stream>>>(w_h1, W2, b2, w_h2);
  k_gemm3    <<<1024, 32, 0, stream>>>(w_h2, W3, b3, gum, o_PL, w_la);

  size_t smem6 = (size_t)(128 * 129 + 128) * sizeof(float) + 3 * 128 * sizeof(int);
  k_sinkhorn <<<32, 128, smem6, stream>>>(w_la, w_pc, w_or);

  k_permW    <<<dim3(128, 32), 128, 0, stream>>>(w_pc, o_L, o_P, o_W);

  size_t smem8 = (size_t)(NN3 + 128 * 129) * sizeof(float) + 128 * sizeof(int);
  k_ancestral<<<dim3(16, 32), 128, smem8, stream>>>(o_L, w_or, w_pc, interv, z_anc, o_QZ);

  k_xrecons  <<<dim3(2, 4096), dim3(32, 8), 0, stream>>>(o_QZ, Wd, o_XR);
}